// gcc_Conv2d_41841571398171
// MI455X (gfx1250) — compile-verified
//
#include <hip/hip_runtime.h>
#include <hip/hip_bf16.h>

typedef __attribute__((ext_vector_type(2))) float v2f;
typedef __attribute__((ext_vector_type(8))) float v8f;

#define BATCH 32
#define CHAN  384
#define S     56
#define SS    (S*S)          // 3136
#define KSTEPS 14            // 56 / 4

// One block = 128 threads = 4 waves; block handles one (c,b) slice.
// Wave w handles w-tile w (columns 16w..16w+15); each wave computes all 4
// 16-row M-tiles (rows 0..63, rows 56..63 discarded) so B is loaded once.
__global__ __launch_bounds__(128)
void circ_conv_wmma_kernel(const float* __restrict__ x,
                           const float* __restrict__ weight,
                           const float* __restrict__ bias,
                           const float* __restrict__ pe,
                           float* __restrict__ out) {
    __shared__ float s_w[168];   // wgt[c, i % 56], i in [0,168): handles (j-row)+112
    __shared__ float s_pe[S];

    const int blk = blockIdx.x;
    const int c = blk / BATCH;
    const int b = blk % BATCH;

    // cooperative LDS fill (all waves share channel c)
    for (int i = threadIdx.x; i < 168; i += 128)
        s_w[i] = weight[c * S + (i % S)];
    if (threadIdx.x < S)
        s_pe[threadIdx.x] = pe[c * S + threadIdx.x];
    __syncthreads();

    const int wave   = threadIdx.x >> 5;   // w-tile id 0..3
    const int lane   = threadIdx.x & 31;
    const int h32    = lane >> 4;          // lane half (selects K pair / M+8)
    const int lanelo = lane & 15;

    const int w      = wave * 16 + lanelo; // output/input column
    const bool wvalid = (w < S);           // only wave 3 has invalid columns

    const float* xc = x + (size_t)(b * CHAN + c) * SS;

    v8f acc0 = {}, acc1 = {}, acc2 = {}, acc3 = {};

    #pragma unroll
    for (int s = 0; s < KSTEPS; ++s) {
        const int j0 = 4 * s + 2 * h32;    // K indices held by this lane half
        const int j1 = j0 + 1;

        // B operand (4x16, K x N): rows j0/j1 at column w, with PE folded in.
        float b0 = 0.f, b1 = 0.f;
        if (wvalid) {
            b0 = xc[j0 * S + w] + s_pe[j0];
            b1 = xc[j1 * S + w] + s_pe[j1];
        }
        v2f Bv = { b0, b1 };

        // A operand per M-tile: circulant row (h0+lanelo), cols j0/j1.
        // idx = j - row + 112 stays in [49,167]; rows >=56 read junk that only
        // reaches discarded output rows.
        const int r0 = lanelo;             // mt*16 subtracted via offset below
        {
            v2f Av = { s_w[j0 - r0 + 112],       s_w[j1 - r0 + 112] };
            acc0 = __builtin_amdgcn_wmma_f32_16x16x4_f32(false, Av, false, Bv,
                                                         (short)0, acc0, false, false);
        }
        {
            v2f Av = { s_w[j0 - r0 - 16 + 112],  s_w[j1 - r0 - 16 + 112] };
            acc1 = __builtin_amdgcn_wmma_f32_16x16x4_f32(false, Av, false, Bv,
                                                         (short)0, acc1, false, false);
        }
        {
            v2f Av = { s_w[j0 - r0 - 32 + 112],  s_w[j1 - r0 - 32 + 112] };
            acc2 = __builtin_amdgcn_wmma_f32_16x16x4_f32(false, Av, false, Bv,
                                                         (short)0, acc2, false, false);
        }
        {
            v2f Av = { s_w[j0 - r0 - 48 + 112],  s_w[j1 - r0 - 48 + 112] };
            acc3 = __builtin_amdgcn_wmma_f32_16x16x4_f32(false, Av, false, Bv,
                                                         (short)0, acc3, false, false);
        }
    }

    // Store: C/D layout — VGPR i holds row (h0 + i + 8*h32), col = w.
    const float bi = bias[c];
    float* oc = out + (size_t)(b * CHAN + c) * SS;
    if (wvalid) {
        #pragma unroll
        for (int i = 0; i < 8; ++i) {
            int r;
            r = 0  + i + 8 * h32;             oc[r * S + w] = acc0[i] + bi;
            r = 16 + i + 8 * h32;             oc[r * S + w] = acc1[i] + bi;
            r = 32 + i + 8 * h32;             oc[r * S + w] = acc2[i] + bi;
            r = 48 + i + 8 * h32;
            if (r < S)                        oc[r * S + w] = acc3[i] + bi;
        }
    }
}

extern "C" void kernel_launch(void* const* d_in, const int* in_sizes, int n_in,
                              void* d_out, int out_size, void* d_ws, size_t ws_size,
                              hipStream_t stream) {
    const float* x      = (const float*)d_in[0];  // [32,384,56,56]
    const float* weight = (const float*)d_in[1];  // [384,1,56,1]
    const float* bias   = (const float*)d_in[2];  // [384]
    const float* pe     = (const float*)d_in[3];  // [1,384,56,1]
    float* out          = (float*)d_out;          // [32,384,56,56]

    dim3 grid(CHAN * BATCH);  // 12288 blocks
    dim3 block(128);          // 4 waves
    circ_conv_wmma_kernel<<<grid, block, 0, stream>>>(x, weight, bias, pe, out);
}